// TopoBrainPhysical_7593502179678
// MI455X (gfx1250) — compile-verified
//
#include <hip/hip_runtime.h>
#include <math.h>

// ---------------------------------------------------------------------------
// TopoBrainPhysical on MI455X (gfx1250, wave32).
// All dense layers run as D = W^T x X on V_WMMA_F32_16X16X4_F32 (full f32),
// samples in the N dimension (lane%16 = sample, two sample tiles per wave).
// Layers are chained IN D-LAYOUT: consumer kstep kk uses producer D regs
// (D[2kk], D[2kk+1]) directly as (Bv0, Bv1); the weight A-operands in LDS are
// pre-permuted so contraction slot 4kk+c maps to producer feature
// 16*(kk/4) + 2*(kk%4) + (c&1) + 8*(c>>1).  -> near-zero cross-lane shuffles.
// ---------------------------------------------------------------------------

typedef __attribute__((ext_vector_type(2))) float v2f;
typedef __attribute__((ext_vector_type(8))) float v8f;

#define MSG_ANG 4
#define MSG_RAD 2
#define PI_F 3.14159274101257324f

#if __has_builtin(__builtin_amdgcn_tanhf)
#define TANHF(x) __builtin_amdgcn_tanhf(x)
#elif __has_builtin(__builtin_amdgcn_tanh_f32)
#define TANHF(x) __builtin_amdgcn_tanh_f32(x)
#else
#define TANHF(x) tanhf(x)
#endif

#define WMMA4(A, B, C) \
  __builtin_amdgcn_wmma_f32_16x16x4_f32(false, (A), false, (B), (short)0, (C), false, false)

__device__ __forceinline__ float xhalf(float v) { return __shfl_xor(v, 16, 32); }

// Build one B-operand VGPR for sample-tile t from per-lane (feature-major) regs.
__device__ __forceinline__ float bopnd(float flo, float fhi, int t, int lane) {
  if (t == 0) { float m = xhalf(fhi); return (lane < 16) ? flo : m; }
  else        { float m = xhalf(flo); return (lane < 16) ? m : fhi; }
}
// Fold D regs of both sample tiles to a per-lane (feature-major) register.
__device__ __forceinline__ float ffold(float d0, float d1, int lane) {
  float m = xhalf(d1); return (lane < 16) ? d0 : m;
}

// ---- LDS staging layout (float offsets), all A tiles are 64 floats --------
constexpr int A_E1 = 0;               // enc_w1 : MT2*KS1  = 2 tiles (identity)
constexpr int A_E2 = A_E1 + 2 * 64;   // enc_w2 : MT2*KS8  = 16 tiles (chain)
constexpr int A_E3 = A_E2 + 16 * 64;  // enc_w3 : MT1*KS8  = 8  tiles (chain)
constexpr int A_N1 = A_E3 + 8 * 64;   // node_w1: MT2*KS1  = 2  tiles (identity,K=4)
constexpr int A_N2 = A_N1 + 2 * 64;   // node_w2: MT1*KS8  = 8  tiles (chain)
constexpr int A_R1 = A_N2 + 8 * 64;   // ro_w1  : MT2*KS32 = 64 tiles (chain,RB=12)
constexpr int A_R2 = A_R1 + 64 * 64;  // ro_w2  : MT1*KS8  = 8  tiles (chain)
constexpr int A_END = A_R2 + 8 * 64;
constexpr int B_E1 = A_END;           // biases padded to MT*16
constexpr int B_E2 = B_E1 + 32;
constexpr int B_E3 = B_E2 + 32;
constexpr int B_N1 = B_E3 + 16;
constexpr int B_N2 = B_N1 + 32;
constexpr int B_R1 = B_N2 + 16;
constexpr int B_R2 = B_R1 + 32;
constexpr int LDS_TOT = B_R2 + 16;    // 7088 floats = 28352 bytes

// Stage one layer's weights as pre-permuted A-operand tiles + padded bias.
// CHAIN=false: contraction slot s -> W row s (identity, entry layers).
// CHAIN=true : slot 4kk+c -> W row RB*(kk/4) + e,  e = 2*(kk%4)+(c&1)+8*(c>>1),
//              valid iff e < EB and row < K.   Branch-free fill, once per block.
template <int K, int N, int KS, bool CHAIN, int RB = 16, int EB = 16>
__device__ __forceinline__ void fill_weights(float* __restrict__ smem,
                                             int aoff, int boff,
                                             const float* __restrict__ W,
                                             const float* __restrict__ bias,
                                             int tid) {
  constexpr int MT = (N + 15) / 16;
  for (int e = tid; e < MT * KS * 64; e += 256) {
    int tile = e >> 6;                  // tm*KS + kk
    int tm = tile / KS, kk = tile - tm * KS;
    int lane = (e >> 1) & 31, comp = e & 1;
    int m = tm * 16 + (lane & 15);
    int c = ((lane & 16) ? 2 : 0) + comp;
    int f, ok;
    if (CHAIN) {
      int j = kk & 3, blk = kk >> 2;
      int ei = 2 * j + (c & 1) + 8 * (c >> 1);
      f = RB * blk + ei;
      ok = (ei < EB) && (f < K);
    } else {
      f = kk * 4 + c;
      ok = (f < K);
    }
    ok = ok && (m < N);
    int fi = (f < K) ? f : 0;
    int mi = (m < N) ? m : 0;
    float v = W[fi * N + mi];
    smem[aoff + e] = ok ? v : 0.0f;
  }
  for (int e = tid; e < MT * 16; e += 256) {
    smem[boff + e] = (e < N) ? bias[e] : 0.0f;
  }
}

__device__ __forceinline__ v8f bias8(const float* __restrict__ Bsm, int tm, int lane) {
  const float* bp = Bsm + tm * 16 + ((lane >= 16) ? 8 : 0);
  float4 b0 = *(const float4*)bp;
  float4 b1 = *(const float4*)(bp + 4);
  v8f a;
  a[0] = b0.x; a[1] = b0.y; a[2] = b0.z; a[3] = b0.w;
  a[4] = b1.x; a[5] = b1.y; a[6] = b1.z; a[7] = b1.w;
  return a;
}

// Entry layer (KS=1): pre-built B operands (identity slot map), D-layout out.
template <int N, bool DO_TANH>
__device__ __forceinline__ void layerB(const float* __restrict__ Asm,
                                       const float* __restrict__ Bsm,
                                       v2f B0, v2f B1,
                                       float* d0, float* d1, int lane) {
  constexpr int MT = (N + 15) / 16;
#pragma unroll
  for (int tm = 0; tm < MT; ++tm) {
    v8f a0 = bias8(Bsm, tm, lane);
    v8f a1 = a0;
    v2f A = *(const v2f*)(Asm + tm * 64 + lane * 2);
    a0 = WMMA4(A, B0, a0);
    a1 = WMMA4(A, B1, a1);
#pragma unroll
    for (int g = 0; g < 8; ++g) {
      d0[tm * 8 + g] = DO_TANH ? TANHF(a0[g]) : a0[g];
      d1[tm * 8 + g] = DO_TANH ? TANHF(a1[g]) : a1[g];
    }
  }
}

// Chained layer: consumes producer D regs directly (din has 2*KS regs/stile).
template <int KS, int N, bool DO_TANH>
__device__ __forceinline__ void layerD(const float* __restrict__ Asm,
                                       const float* __restrict__ Bsm,
                                       const float* din0, const float* din1,
                                       float* d0, float* d1, int lane) {
  constexpr int MT = (N + 15) / 16;
#pragma unroll
  for (int tm = 0; tm < MT; ++tm) {
    v8f a0 = bias8(Bsm, tm, lane);
    v8f a1 = a0;
#pragma unroll
    for (int kk = 0; kk < KS; ++kk) {
      v2f A = *(const v2f*)(Asm + (tm * KS + kk) * 64 + lane * 2);
      v2f B0; B0.x = din0[2 * kk]; B0.y = din0[2 * kk + 1];
      v2f B1; B1.x = din1[2 * kk]; B1.y = din1[2 * kk + 1];
      a0 = WMMA4(A, B0, a0);
      a1 = WMMA4(A, B1, a1);
    }
#pragma unroll
    for (int g = 0; g < 8; ++g) {
      d0[tm * 8 + g] = DO_TANH ? TANHF(a0[g]) : a0[g];
      d1[tm * 8 + g] = DO_TANH ? TANHF(a1[g]) : a1[g];
    }
  }
}

__global__ __launch_bounds__(256) void topo_brain_kernel(
    const float* __restrict__ x,
    const float* __restrict__ ew1, const float* __restrict__ eb1,
    const float* __restrict__ ew2, const float* __restrict__ eb2,
    const float* __restrict__ ew3, const float* __restrict__ eb3,
    const float* __restrict__ nw1, const float* __restrict__ nb1,
    const float* __restrict__ nw2, const float* __restrict__ nb2,
    const float* __restrict__ angl, const float* __restrict__ radl,
    const float* __restrict__ rw1, const float* __restrict__ rb1,
    const float* __restrict__ rw2, const float* __restrict__ rb2,
    float* __restrict__ out, int L) {
  __shared__ __align__(16) float smem[LDS_TOT];
  const int tid = threadIdx.x;
  const int lane = tid & 31;
  const int wave = tid >> 5;
  const long long sb = (long long)blockIdx.x * 256 + wave * 32; // wave sample base
  (void)radl; // radial mixing reduces exactly to a half-swap (r == [[0,1],[1,0]])

  // stage weights once per block
  fill_weights<4, 24, 1, false>(smem, A_E1, B_E1, ew1, eb1, tid);
  fill_weights<24, 24, 8, true>(smem, A_E2, B_E2, ew2, eb2, tid);
  fill_weights<24, 3, 8, true>(smem, A_E3, B_E3, ew3, eb3, tid);
  fill_weights<4, 24, 1, false>(smem, A_N1, B_N1, nw1, nb1, tid); // embeds 4..11 are 0
  fill_weights<24, 12, 8, true>(smem, A_N2, B_N2, nw2, nb2, tid);
  fill_weights<96, 24, 32, true, 12, 12>(smem, A_R1, B_R1, rw1, rb1, tid);
  fill_weights<24, 4, 8, true>(smem, A_R2, B_R2, rw2, rb2, tid);
  __syncthreads();

  // ---- enc1 input loaded from global DIRECTLY in B-operand layout ----
  const int ln15 = lane & 15;
  const int koff = (lane & 16) ? 2 : 0; // k-rows (0,1) lo-half, (2,3) hi-half
  v2f Bx0 = *(const v2f*)(x + (sb + ln15) * (long long)(L * 4) + (L - 1) * 4 + koff);
  v2f Bx1 = *(const v2f*)(x + (sb + 16 + ln15) * (long long)(L * 4) + (L - 1) * 4 + koff);

  // ---- encoder chain (D-layout all the way) ----
  float e1d0[16], e1d1[16];
  layerB<24, true>(smem + A_E1, smem + B_E1, Bx0, Bx1, e1d0, e1d1, lane);
  float e2d0[16], e2d1[16];
  layerD<8, 24, true>(smem + A_E2, smem + B_E2, e1d0, e1d1, e2d0, e2d1, lane);
  float e3d0[8], e3d1[8];
  layerD<8, 3, false>(smem + A_E3, smem + B_E3, e2d0, e2d1, e3d0, e3d1, lane);
  float z0 = ffold(e3d0[0], e3d1[0], lane);
  float z1 = ffold(e3d0[1], e3d1[1], lane);
  float z2 = ffold(e3d0[2], e3d1[2], lane);

  // ---- per-sample bilinear scatter (pure per-lane) ----
  float Ridx = (1.0f / (1.0f + __expf(-z0))) * (float)(MSG_RAD - 1);
  float phi = (z1 + PI_F) * ((float)MSG_ANG / (2.0f * PI_F));
  float r0 = truncf(Ridx), p0 = truncf(phi);
  float dr = Ridx - r0, dp = phi - p0;
  int r0i = (int)r0, p0i = (int)p0;
  float sm[8], sw[8];
#pragma unroll
  for (int n = 0; n < 8; ++n) { sm[n] = 0.0f; sw[n] = 0.0f; }
#pragma unroll
  for (int c = 0; c < 4; ++c) {
    int dro = c >> 1, dpo = c & 1;
    float wr = dro ? dr : (1.0f - dr);
    float wp = dpo ? dp : (1.0f - dp);
    float w = wr * wp;
    float mk = (w > 0.0f) ? 1.0f : 0.0f;
    int ri = r0i + dro; ri = (ri < (MSG_RAD - 1)) ? ri : (MSG_RAD - 1);
    int pi = (p0i + dpo) % MSG_ANG; if (pi < 0) pi += MSG_ANG;
    int cell = ri * MSG_ANG + pi;
#pragma unroll
    for (int n = 0; n < 8; ++n) {
      float hit = (cell == n) ? mk : 0.0f;
      sm[n] += hit;
      sw[n] += hit * w;
    }
  }

  // ---- graph mixing: h + h_ang + h_rad ----
  float ea[4] = { __expf(angl[0]), __expf(angl[1]), __expf(angl[2]), __expf(angl[3]) };
  float am[4], ap[4];
#pragma unroll
  for (int i = 0; i < 4; ++i) {
    float mm = ea[(i + 3) & 3], pp = ea[(i + 1) & 3];
    float d = mm + pp; d = (d > 1e-6f) ? d : 1e-6f;
    am[i] = mm / d; ap[i] = pp / d;
  }
  float S2[8], W2[8];
#pragma unroll
  for (int n = 0; n < 8; ++n) {
    int i = n >> 1, p = n & 1;
    int nm = (((i + 3) & 3) << 1) | p;
    int np = (((i + 1) & 3) << 1) | p;
    int nr = (n + 4) & 7;
    S2[n] = sm[n] + am[i] * sm[nm] + ap[i] * sm[np] + sm[nr];
    W2[n] = sw[n] + am[i] * sw[nm] + ap[i] * sw[np] + sw[nr];
  }

  // ---- node MLP + readout-1 accumulation (node-by-node, D-chained) ----
  v8f r00 = bias8(smem + B_R1, 0, lane); // stile0, tm0
  v8f r01 = bias8(smem + B_R1, 1, lane); // stile0, tm1
  v8f r10 = r00, r11 = r01;              // stile1
#pragma unroll
  for (int n = 0; n < 8; ++n) {
    float h0 = S2[n] * z0, h1 = S2[n] * z1, h2 = S2[n] * z2, h3 = W2[n];
    v2f Bn0, Bn1;
    Bn0.x = bopnd(h0, h2, 0, lane); Bn0.y = bopnd(h1, h3, 0, lane);
    Bn1.x = bopnd(h0, h2, 1, lane); Bn1.y = bopnd(h1, h3, 1, lane);
    float n1d0[16], n1d1[16];
    layerB<24, true>(smem + A_N1, smem + B_N1, Bn0, Bn1, n1d0, n1d1, lane);
    float n2d0[8], n2d1[8];
    layerD<8, 12, false>(smem + A_N2, smem + B_N2, n1d0, n1d1, n2d0, n2d1, lane);
#pragma unroll
    for (int j = 0; j < 4; ++j) {
      int kk = n * 4 + j;
      v2f Bq0; Bq0.x = n2d0[2 * j]; Bq0.y = n2d0[2 * j + 1];
      v2f Bq1; Bq1.x = n2d1[2 * j]; Bq1.y = n2d1[2 * j + 1];
      v2f A0 = *(const v2f*)(smem + A_R1 + (0 * 32 + kk) * 64 + lane * 2);
      v2f A1 = *(const v2f*)(smem + A_R1 + (1 * 32 + kk) * 64 + lane * 2);
      r00 = WMMA4(A0, Bq0, r00);
      r10 = WMMA4(A0, Bq1, r10);
      r01 = WMMA4(A1, Bq0, r01);
      r11 = WMMA4(A1, Bq1, r11);
    }
  }

  // ---- readout tanh + final layer ----
  float q0[16], q1[16];
#pragma unroll
  for (int g = 0; g < 8; ++g) {
    q0[g] = TANHF(r00[g]); q0[8 + g] = TANHF(r01[g]);
    q1[g] = TANHF(r10[g]); q1[8 + g] = TANHF(r11[g]);
  }
  float o0[8], o1[8];
  layerD<8, 4, false>(smem + A_R2, smem + B_R2, q0, q1, o0, o1, lane);
  float O0 = ffold(o0[0], o1[0], lane);
  float O1 = ffold(o0[1], o1[1], lane);
  float O2 = ffold(o0[2], o1[2], lane);
  float O3 = ffold(o0[3], o1[3], lane);

  *(float4*)(out + (sb + lane) * 4) = make_float4(O0, O1, O2, O3);
}

extern "C" void kernel_launch(void* const* d_in, const int* in_sizes, int n_in,
                              void* d_out, int out_size, void* d_ws, size_t ws_size,
                              hipStream_t stream) {
  (void)n_in; (void)d_ws; (void)ws_size;
  const float* x    = (const float*)d_in[0];
  const float* ew1  = (const float*)d_in[1];
  const float* eb1  = (const float*)d_in[2];
  const float* ew2  = (const float*)d_in[3];
  const float* eb2  = (const float*)d_in[4];
  const float* ew3  = (const float*)d_in[5];
  const float* eb3  = (const float*)d_in[6];
  const float* nw1  = (const float*)d_in[7];
  const float* nb1  = (const float*)d_in[8];
  const float* nw2  = (const float*)d_in[9];
  const float* nb2  = (const float*)d_in[10];
  const float* angl = (const float*)d_in[11];
  const float* radl = (const float*)d_in[12];
  const float* rw1  = (const float*)d_in[13];
  const float* rb1  = (const float*)d_in[14];
  const float* rw2  = (const float*)d_in[15];
  const float* rb2  = (const float*)d_in[16];
  float* out = (float*)d_out;

  const int B = out_size / 4;          // (B,4) output
  const int L = in_sizes[0] / (B * 4); // x is (B,L,4)
  const int blocks = (B + 255) / 256;  // B = 524288 -> 2048 blocks, 8 waves each

  topo_brain_kernel<<<blocks, 256, 0, stream>>>(
      x, ew1, eb1, ew2, eb2, ew3, eb3, nw1, nb1, nw2, nb2,
      angl, radl, rw1, rb1, rw2, rb2, out, L);
}